// GCN_24919400251445
// MI455X (gfx1250) — compile-verified
//
#include <hip/hip_runtime.h>
#include <hip/hip_bf16.h>
#include <math.h>

typedef __attribute__((ext_vector_type(2))) float v2f;
typedef __attribute__((ext_vector_type(8))) float v8f;

// ---------------------------------------------------------------------------
// Degree / normalization kernels
// ---------------------------------------------------------------------------
__global__ void deg_init_kernel(float* __restrict__ deg, int N) {
    int i = blockIdx.x * blockDim.x + threadIdx.x;
    if (i < N) deg[i] = 1.0f;   // self-loop contributes 1
}

__global__ void deg_count_kernel(const int* __restrict__ dst, float* __restrict__ deg, int E) {
    int e = blockIdx.x * blockDim.x + threadIdx.x;
    if (e < E) atomicAdd(&deg[dst[e]], 1.0f);
}

__global__ void dinv_kernel(float* __restrict__ deg, int N) {
    int i = blockIdx.x * blockDim.x + threadIdx.x;
    if (i < N) deg[i] = rsqrtf(deg[i]);   // deg >= 1 always (self-loop)
}

// ---------------------------------------------------------------------------
// Dense transform via V_WMMA_F32_16X16X4_F32:  t = h_in @ W  (fp32 end-to-end)
// Writes s = dinv * t into both `s` (gather source) and `acc` (self-loop init).
// One wave per 16-node tile. K padded to a multiple of 4 with *branch-free*
// zero padding (index clamp + 0/1 mask -> v_cndmask, no EXEC divergence, so
// the WMMA sequence runs with EXEC all-ones throughout).
// A layout: lane L (m = L&15, kk = (L>>4)*2): {h[m][k0+kk], h[m][k0+kk+1]}
// B layout: lane L (n = L&15):                {W[k0+kk][n], W[k0+kk+1][n]}
// C/D layout: VGPR r -> node (r + 8*(L>>4)), col (L&15)
// ---------------------------------------------------------------------------
template <int K, int DOUT>
__global__ void transform_wmma_kernel(const float* __restrict__ hin,
                                      const float* __restrict__ W,     // [K][DOUT]
                                      const float* __restrict__ dinv,
                                      float* __restrict__ s,
                                      float* __restrict__ acc,
                                      int n_tiles) {
    int wave = (blockIdx.x * blockDim.x + threadIdx.x) >> 5;
    if (wave >= n_tiles) return;               // wave-uniform guard: EXEC stays all-ones
    int lane = threadIdx.x & 31;
    int m  = lane & 15;
    int kk = (lane >> 4) * 2;
    int node_a = wave * 16 + m;

    v8f c0 = {};
    v8f c1 = {};
    constexpr int NK = (K + 3) / 4;
#pragma unroll
    for (int ks = 0; ks < NK; ++ks) {
        int k0 = ks * 4;
        int ka = k0 + kk, kb = k0 + kk + 1;
        // branch-free zero padding (only instantiated for K not divisible by 4)
        int   kac = (ka < K) ? ka : (K - 1);
        int   kbc = (kb < K) ? kb : (K - 1);
        float ma  = (ka < K) ? 1.0f : 0.0f;
        float mb  = (kb < K) ? 1.0f : 0.0f;

        v2f a;
        a.x = hin[(size_t)node_a * K + kac] * ma;
        a.y = hin[(size_t)node_a * K + kbc] * mb;
        v2f b0;
        b0.x = W[kac * DOUT + m] * ma;
        b0.y = W[kbc * DOUT + m] * mb;
        c0 = __builtin_amdgcn_wmma_f32_16x16x4_f32(false, a, false, b0,
                                                   (short)0, c0, false, false);
        if (DOUT == 32) {
            v2f b1;
            b1.x = W[kac * DOUT + 16 + m] * ma;
            b1.y = W[kbc * DOUT + 16 + m] * mb;
            c1 = __builtin_amdgcn_wmma_f32_16x16x4_f32(false, a, false, b1,
                                                       (short)0, c1, false, false);
        }
    }

    int hi = lane >> 4;
#pragma unroll
    for (int r = 0; r < 8; ++r) {
        int node = wave * 16 + r + 8 * hi;
        float dv = dinv[node];
        {
            float v = dv * c0[r];
            size_t idx = (size_t)node * DOUT + m;
            s[idx] = v;  acc[idx] = v;
        }
        if (DOUT == 32) {
            float v = dv * c1[r];
            size_t idx = (size_t)node * DOUT + 16 + m;
            s[idx] = v;  acc[idx] = v;
        }
    }
}

// ---------------------------------------------------------------------------
// Edge scatter-add: acc[dst] += s[src].  One lane per edge; float4 gathers,
// per-float f32 atomics (32 lanes hit 32 node rows -> spread over L2 channels).
// ---------------------------------------------------------------------------
template <int DOUT>
__global__ void aggregate_kernel(const int* __restrict__ src, const int* __restrict__ dst,
                                 const float* __restrict__ s, float* __restrict__ acc, int E) {
    int e = blockIdx.x * blockDim.x + threadIdx.x;
    if (e >= E) return;
    int si = src[e], di = dst[e];
    const float4* sv = (const float4*)(s + (size_t)si * DOUT);
    float* ap = acc + (size_t)di * DOUT;
#pragma unroll
    for (int c = 0; c < DOUT / 4; ++c) {
        float4 v = sv[c];
        atomicAdd(ap + 4 * c + 0, v.x);
        atomicAdd(ap + 4 * c + 1, v.y);
        atomicAdd(ap + 4 * c + 2, v.z);
        atomicAdd(ap + 4 * c + 3, v.w);
    }
}

// ---------------------------------------------------------------------------
// Finalize: h_out[n][f] = act(dinv[n] * acc[n][f] + b[f])
// ---------------------------------------------------------------------------
template <int DOUT, bool RELU>
__global__ void finalize_kernel(const float* __restrict__ acc, const float* __restrict__ dinv,
                                const float* __restrict__ bias, float* __restrict__ hout, int N) {
    int n = blockIdx.x * blockDim.x + threadIdx.x;
    if (n >= N) return;
    float dv = dinv[n];
    const float* ar = acc + (size_t)n * DOUT;
    float* hr = hout + (size_t)n * DOUT;
#pragma unroll
    for (int f = 0; f < DOUT; ++f) {
        float v = dv * ar[f] + bias[f];
        if (RELU) v = fmaxf(v, 0.0f);
        hr[f] = v;
    }
}

// ---------------------------------------------------------------------------
// Layer 4 (32 -> 1): VALU dot product; a WMMA tile would be 15/16 wasted.
// ---------------------------------------------------------------------------
__global__ void l4_transform_kernel(const float* __restrict__ h, const float* __restrict__ W4,
                                    const float* __restrict__ dinv, float* __restrict__ s1,
                                    float* __restrict__ acc1, int N) {
    int n = blockIdx.x * blockDim.x + threadIdx.x;
    if (n >= N) return;
    const float4* hr = (const float4*)(h + (size_t)n * 32);
    const float4* wr = (const float4*)W4;
    float t = 0.0f;
#pragma unroll
    for (int c = 0; c < 8; ++c) {
        float4 a = hr[c], w = wr[c];
        t += a.x * w.x + a.y * w.y + a.z * w.z + a.w * w.w;
    }
    float v = dinv[n] * t;
    s1[n] = v;
    acc1[n] = v;
}

__global__ void aggregate1_kernel(const int* __restrict__ src, const int* __restrict__ dst,
                                  const float* __restrict__ s1, float* __restrict__ acc1, int E) {
    int e = blockIdx.x * blockDim.x + threadIdx.x;
    if (e < E) atomicAdd(&acc1[dst[e]], s1[src[e]]);
}

__global__ void final_sigmoid_kernel(const float* __restrict__ acc1, const float* __restrict__ dinv,
                                     const float* __restrict__ b4, float* __restrict__ out, int N) {
    int n = blockIdx.x * blockDim.x + threadIdx.x;
    if (n >= N) return;
    float z = dinv[n] * acc1[n] + b4[0];
    out[n] = 1.0f / (1.0f + __expf(-z));
}

// ---------------------------------------------------------------------------
// Host launcher
// ---------------------------------------------------------------------------
extern "C" void kernel_launch(void* const* d_in, const int* in_sizes, int n_in,
                              void* d_out, int out_size, void* d_ws, size_t ws_size,
                              hipStream_t stream) {
    (void)n_in; (void)out_size; (void)ws_size;
    const float* x   = (const float*)d_in[0];
    const int*   ei  = (const int*)d_in[1];
    const float* W1  = (const float*)d_in[2];
    const float* b1  = (const float*)d_in[3];
    const float* W2  = (const float*)d_in[4];
    const float* b2  = (const float*)d_in[5];
    const float* W3  = (const float*)d_in[6];
    const float* b3  = (const float*)d_in[7];
    const float* W4  = (const float*)d_in[8];
    const float* b4  = (const float*)d_in[9];
    float* out = (float*)d_out;

    const int N = in_sizes[0] / 23;
    const int E = in_sizes[1] / 2;
    const int* src = ei;
    const int* dst = ei + E;

    // Workspace: dinv[N] | s[32N] | acc[32N] | h[32N]  (~77.6 MB, L2-resident)
    float* dinv = (float*)d_ws;
    float* s    = dinv + (size_t)N;
    float* acc  = s    + (size_t)N * 32;
    float* h    = acc  + (size_t)N * 32;

    const int TB = 256;
    const int gN = (N + TB - 1) / TB;
    const int gE = (E + TB - 1) / TB;
    const int n_tiles = (N + 15) / 16;              // N=200000 -> 12500 exact
    const int gT = (n_tiles * 32 + 127) / 128;      // 4 waves per block

    // --- normalization ---
    deg_init_kernel<<<gN, TB, 0, stream>>>(dinv, N);
    deg_count_kernel<<<gE, TB, 0, stream>>>(dst, dinv, E);
    dinv_kernel<<<gN, TB, 0, stream>>>(dinv, N);

    // --- layer 1: 23 -> 16, relu ---
    transform_wmma_kernel<23, 16><<<gT, 128, 0, stream>>>(x, W1, dinv, s, acc, n_tiles);
    aggregate_kernel<16><<<gE, TB, 0, stream>>>(src, dst, s, acc, E);
    finalize_kernel<16, true><<<gN, TB, 0, stream>>>(acc, dinv, b1, h, N);

    // --- layer 2: 16 -> 32, relu ---
    transform_wmma_kernel<16, 32><<<gT, 128, 0, stream>>>(h, W2, dinv, s, acc, n_tiles);
    aggregate_kernel<32><<<gE, TB, 0, stream>>>(src, dst, s, acc, E);
    finalize_kernel<32, true><<<gN, TB, 0, stream>>>(acc, dinv, b2, h, N);

    // --- layer 3: 32 -> 32, no act ---
    transform_wmma_kernel<32, 32><<<gT, 128, 0, stream>>>(h, W3, dinv, s, acc, n_tiles);
    aggregate_kernel<32><<<gE, TB, 0, stream>>>(src, dst, s, acc, E);
    finalize_kernel<32, false><<<gN, TB, 0, stream>>>(acc, dinv, b3, h, N);

    // --- layer 4: 32 -> 1, sigmoid ---  (reuse first N floats of s/acc)
    l4_transform_kernel<<<gN, TB, 0, stream>>>(h, W4, dinv, s, acc, N);
    aggregate1_kernel<<<gE, TB, 0, stream>>>(src, dst, s, acc, E);
    final_sigmoid_kernel<<<gN, TB, 0, stream>>>(acc, dinv, b4, out, N);
}